// AttnDecoderRNN_24352464568501
// MI455X (gfx1250) — compile-verified
//
#include <hip/hip_runtime.h>
#include <math.h>

#define HSZ 1024
#define VSZ 50257
#define SSZ 4096

typedef __attribute__((ext_vector_type(2))) float v2f;
typedef __attribute__((ext_vector_type(8))) float v8f;

// ---------------------------------------------------------------------------
// K1: gi = w_ih @ x + b_ih ; gh = w_hh @ h_prev + b_hh   (rows 0..3071)
// one wave32 per row, float4 streaming, shfl reduction
// ---------------------------------------------------------------------------
__global__ void k_gates(const int* __restrict__ word, const float* __restrict__ hprev,
                        const float* __restrict__ emb,
                        const float* __restrict__ w_ih, const float* __restrict__ w_hh,
                        const float* __restrict__ b_ih, const float* __restrict__ b_hh,
                        float* __restrict__ gi, float* __restrict__ gh) {
  int wave = threadIdx.x >> 5;
  int lane = threadIdx.x & 31;
  int row  = blockIdx.x * 8 + wave;               // 384 blocks * 8 = 3072 rows
  const float4* x  = (const float4*)(emb + (size_t)word[0] * HSZ);
  const float4* hv = (const float4*)hprev;
  const float4* wi = (const float4*)(w_ih + (size_t)row * HSZ);
  const float4* wh = (const float4*)(w_hh + (size_t)row * HSZ);
  float si = 0.f, sh = 0.f;
  #pragma unroll
  for (int i = lane; i < HSZ / 4; i += 32) {
    float4 a = wi[i], b = x[i];
    si += a.x * b.x + a.y * b.y + a.z * b.z + a.w * b.w;
    float4 c = wh[i], d = hv[i];
    sh += c.x * d.x + c.y * d.y + c.z * d.z + c.w * d.w;
  }
  for (int off = 16; off; off >>= 1) {
    si += __shfl_down(si, off, 32);
    sh += __shfl_down(sh, off, 32);
  }
  if (lane == 0) {
    gi[row] = si + b_ih[row];
    gh[row] = sh + b_hh[row];
  }
}

// ---------------------------------------------------------------------------
// K2: GRU elementwise -> h ; also c = dot(attn_b, h)  (1 block, 1024 thr)
// ---------------------------------------------------------------------------
__global__ void k_gru(const float* __restrict__ gi, const float* __restrict__ gh,
                      const float* __restrict__ hprev, const float* __restrict__ attn_b,
                      float* __restrict__ h_ws, float* __restrict__ c_ws,
                      float* __restrict__ out_hidden) {
  __shared__ float red[32];
  int i = threadIdx.x;
  float r = 1.f / (1.f + __expf(-(gi[i] + gh[i])));
  float z = 1.f / (1.f + __expf(-(gi[HSZ + i] + gh[HSZ + i])));
  float n = tanhf(gi[2 * HSZ + i] + r * gh[2 * HSZ + i]);
  float h = (1.f - z) * n + z * hprev[i];
  h_ws[i] = h;
  out_hidden[i] = h;
  float v = attn_b[i] * h;
  for (int off = 16; off; off >>= 1) v += __shfl_down(v, off, 32);
  if ((i & 31) == 0) red[i >> 5] = v;
  __syncthreads();
  if (i < 32) {
    float v2 = red[i];
    for (int off = 16; off; off >>= 1) v2 += __shfl_down(v2, off, 32);
    if (i == 0) c_ws[0] = v2;
  }
}

// ---------------------------------------------------------------------------
// K3: tpart[jc][k] = sum_{j in chunk jc} attn_w[j][k] * h[j]
// 32 blocks = 4 col-tiles x 8 j-chunks (coalesced attn_w rows)
// ---------------------------------------------------------------------------
__global__ void k_tpart(const float* __restrict__ attn_w, const float* __restrict__ h,
                        float* __restrict__ tpart) {
  __shared__ float hs[128];
  int c  = blockIdx.x & 3;
  int jc = blockIdx.x >> 2;
  int col = c * 256 + threadIdx.x;
  if (threadIdx.x < 128) hs[threadIdx.x] = h[jc * 128 + threadIdx.x];
  __syncthreads();
  const float* wp = attn_w + (size_t)(jc * 128) * HSZ + col;
  float acc = 0.f;
  #pragma unroll 4
  for (int j = 0; j < 128; j++) acc += wp[(size_t)j * HSZ] * hs[j];
  tpart[jc * HSZ + col] = acc;
}

// K3b: t[k] = sum over 8 chunks
__global__ void k_treduce(const float* __restrict__ tpart, float* __restrict__ t) {
  int col = blockIdx.x * 256 + threadIdx.x;  // 4 blocks
  float acc = 0.f;
  #pragma unroll
  for (int jc = 0; jc < 8; jc++) acc += tpart[jc * HSZ + col];
  t[col] = acc;
}

// ---------------------------------------------------------------------------
// K4: scores[s] = dot(enc[s,:], t) + c    one wave per s, 512 blocks
// ---------------------------------------------------------------------------
__global__ void k_scores(const float* __restrict__ enc, const float* __restrict__ t,
                         const float* __restrict__ c_ws, float* __restrict__ scores) {
  int wave = threadIdx.x >> 5, lane = threadIdx.x & 31;
  int s = blockIdx.x * 8 + wave;
  const float4* e  = (const float4*)(enc + (size_t)s * HSZ);
  const float4* tv = (const float4*)t;
  float acc = 0.f;
  #pragma unroll
  for (int i = lane; i < HSZ / 4; i += 32) {
    float4 a = e[i], b = tv[i];
    acc += a.x * b.x + a.y * b.y + a.z * b.z + a.w * b.w;
  }
  for (int off = 16; off; off >>= 1) acc += __shfl_down(acc, off, 32);
  if (lane == 0) scores[s] = acc + c_ws[0];
}

// ---------------------------------------------------------------------------
// K5: softmax over 4096 scores (1 block, 1024 thr, 4 elems/thr)
// ---------------------------------------------------------------------------
__global__ void k_softmax(const float* __restrict__ scores, float* __restrict__ out_attn,
                          float* __restrict__ aw) {
  __shared__ float red[32];
  __shared__ float stat[2];
  int tid = threadIdx.x, lane = tid & 31, wid = tid >> 5;
  float v[4];
  float m = -3.402823e38f;
  #pragma unroll
  for (int k = 0; k < 4; k++) { v[k] = scores[tid + k * 1024]; m = fmaxf(m, v[k]); }
  for (int off = 16; off; off >>= 1) m = fmaxf(m, __shfl_down(m, off, 32));
  if (lane == 0) red[wid] = m;
  __syncthreads();
  if (wid == 0) {
    float mm = red[lane];
    for (int off = 16; off; off >>= 1) mm = fmaxf(mm, __shfl_down(mm, off, 32));
    if (lane == 0) stat[0] = mm;
  }
  __syncthreads();
  float gm = stat[0], s = 0.f;
  #pragma unroll
  for (int k = 0; k < 4; k++) { v[k] = __expf(v[k] - gm); s += v[k]; }
  for (int off = 16; off; off >>= 1) s += __shfl_down(s, off, 32);
  if (lane == 0) red[wid] = s;
  __syncthreads();
  if (wid == 0) {
    float ss = red[lane];
    for (int off = 16; off; off >>= 1) ss += __shfl_down(ss, off, 32);
    if (lane == 0) stat[1] = ss;
  }
  __syncthreads();
  float inv = 1.f / stat[1];
  #pragma unroll
  for (int k = 0; k < 4; k++) {
    float r = v[k] * inv;
    out_attn[tid + k * 1024] = r;
    aw[tid + k * 1024] = r;
  }
}

// ---------------------------------------------------------------------------
// K6: context partials: cpart[chunk][k] = sum_{s in 64-row chunk} w[s]*enc[s][k]
// 64 blocks x 256 thr, each thread covers 4 columns
// ---------------------------------------------------------------------------
__global__ void k_cpart(const float* __restrict__ enc, const float* __restrict__ w,
                        float* __restrict__ cpart) {
  __shared__ float wsm[64];
  int chunk = blockIdx.x, tid = threadIdx.x;
  if (tid < 64) wsm[tid] = w[chunk * 64 + tid];
  __syncthreads();
  float a0 = 0.f, a1 = 0.f, a2 = 0.f, a3 = 0.f;
  const float* e = enc + (size_t)chunk * 64 * HSZ;
  #pragma unroll 2
  for (int s = 0; s < 64; s++) {
    float wv = wsm[s];
    const float* row = e + (size_t)s * HSZ;
    a0 += wv * row[tid];
    a1 += wv * row[256 + tid];
    a2 += wv * row[512 + tid];
    a3 += wv * row[768 + tid];
  }
  cpart[chunk * HSZ + tid]       = a0;
  cpart[chunk * HSZ + 256 + tid] = a1;
  cpart[chunk * HSZ + 512 + tid] = a2;
  cpart[chunk * HSZ + 768 + tid] = a3;
}

// K7: reduce partials -> zvec = concat(h, context)   (1 block, 1024 thr)
__global__ void k_zvec(const float* __restrict__ h, const float* __restrict__ cpart,
                       float* __restrict__ zvec) {
  int col = threadIdx.x;
  float acc = 0.f;
  #pragma unroll 4
  for (int ch = 0; ch < 64; ch++) acc += cpart[ch * HSZ + col];
  zvec[col]       = h[col];
  zvec[HSZ + col] = acc;
}

// ---------------------------------------------------------------------------
// K8: output[v] = dot(out_w[v,0:2048], zvec) + out_b[v]  via WMMA f32 16x16x4
// One wave per 16-row tile. A-frag: lanes 0-15 hold rows m=lane (K, K+1),
// lanes 16-31 hold rows m=lane-16 (K+2, K+3) -> each lane one v2f load.
// B-frag: zvec chunk broadcast across all 16 N columns (from LDS), so every
// D column equals the GEMV result; lanes 0-7 store rows 0-7 (d[lane]),
// lanes 16-23 store rows 8-15 (d[lane-16]). EXEC is all-ones around WMMA.
// ---------------------------------------------------------------------------
__global__ void k_out(const float* __restrict__ out_w, const float* __restrict__ out_b,
                      const float* __restrict__ zvec, float* __restrict__ y) {
  __shared__ float zs[2 * HSZ];
  int tid = threadIdx.x;
  #pragma unroll
  for (int i = tid; i < 2 * HSZ; i += 256) zs[i] = zvec[i];
  __syncthreads();

  int wave = tid >> 5, lane = tid & 31;
  int tile = blockIdx.x * 8 + wave;
  int m    = lane & 15;
  int row  = tile * 16 + m;
  int rowc = row < VSZ ? row : VSZ - 1;        // clamp: keep EXEC full for WMMA
  int half = (lane >> 4) << 1;                 // 0 for lanes 0-15, 2 for 16-31
  const float* arow = out_w + (size_t)rowc * (2 * HSZ) + half;
  const float* brow = zs + half;

  v8f acc = {};
  #pragma unroll 8
  for (int k = 0; k < 2 * HSZ; k += 4) {
    v2f a = *(const v2f*)(arow + k);           // 16 parallel row streams / wave
    v2f b = *(const v2f*)(brow + k);           // broadcast vector chunk (LDS)
    acc = __builtin_amdgcn_wmma_f32_16x16x4_f32(
        /*neg_a=*/false, a, /*neg_b=*/false, b,
        /*c_mod=*/(short)0, acc, /*reuse_a=*/false, /*reuse_b=*/false);
  }

  int r = -1;
  if (lane < 8) r = lane;                      // rows 0..7, d[lane]
  else if (lane >= 16 && lane < 24) r = lane - 8;  // rows 8..15, d[lane-16]
  if (r >= 0) {
    int orow = tile * 16 + r;
    if (orow < VSZ) y[orow] = acc[r & 7] + out_b[orow];
  }
}

// ---------------------------------------------------------------------------
extern "C" void kernel_launch(void* const* d_in, const int* in_sizes, int n_in,
                              void* d_out, int out_size, void* d_ws, size_t ws_size,
                              hipStream_t stream) {
  const int*   word   = (const int*)d_in[0];
  const float* lasth  = (const float*)d_in[1];
  const float* enc    = (const float*)d_in[2];
  const float* emb    = (const float*)d_in[3];
  const float* w_ih   = (const float*)d_in[4];
  const float* w_hh   = (const float*)d_in[5];
  const float* b_ih   = (const float*)d_in[6];
  const float* b_hh   = (const float*)d_in[7];
  const float* attn_w = (const float*)d_in[8];
  const float* attn_b = (const float*)d_in[9];
  const float* out_w  = (const float*)d_in[10];
  const float* out_b  = (const float*)d_in[11];
  float* out = (float*)d_out;
  float* ws  = (float*)d_ws;

  // workspace layout (floats)
  float* gi     = ws + 0;       // 3072
  float* gh     = ws + 3072;    // 3072
  float* h      = ws + 6144;    // 1024
  float* cc     = ws + 7168;    // 1 (+pad)
  float* tpart  = ws + 7232;    // 8*1024
  float* t      = ws + 15424;   // 1024
  float* scores = ws + 16448;   // 4096
  float* aw     = ws + 20544;   // 4096
  float* cpart  = ws + 24640;   // 64*1024
  float* zvec   = ws + 90176;   // 2048   (total ~361 KB)

  k_gates  <<<384, 256, 0, stream>>>(word, lasth, emb, w_ih, w_hh, b_ih, b_hh, gi, gh);
  k_gru    <<<1, 1024, 0, stream>>>(gi, gh, lasth, attn_b, h, cc, out + VSZ);
  k_tpart  <<<32, 256, 0, stream>>>(attn_w, h, tpart);
  k_treduce<<<4, 256, 0, stream>>>(tpart, t);
  k_scores <<<512, 256, 0, stream>>>(enc, t, cc, scores);
  k_softmax<<<1, 1024, 0, stream>>>(scores, out + VSZ + HSZ, aw);
  k_cpart  <<<64, 256, 0, stream>>>(enc, aw, cpart);
  k_zvec   <<<1, 1024, 0, stream>>>(h, cpart, zvec);
  k_out    <<<(VSZ + 127) / 128, 256, 0, stream>>>(out_w, out_b, zvec, out);
}